// MuyGPs_layer_57492432224314
// MI455X (gfx1250) — compile-verified
//
#include <hip/hip_runtime.h>
#include <math.h>

#define NF    256   // feature count F
#define NNB   64    // neighbors NN
#define RR    10    // response count R
#define NCOL  11    // RHS columns: 10 targets + Kcross
#define XNS   260   // LDS row stride for xn: 260%64==4 -> conflict-free WMMA frags
#define KS    66    // LDS row stride for K
#define RS    12    // LDS row stride for rhs
#define EPSV  1e-5f
#define SQRT3F 1.7320508075688772f

typedef __attribute__((ext_vector_type(2))) float v2f;
typedef __attribute__((ext_vector_type(8))) float v8f;

__global__ __launch_bounds__(256) void muygps_batch_kernel(
    const float* __restrict__ x,        // [N, F]
    const float* __restrict__ tgt,      // [B, NN, R]
    const float* __restrict__ ls_ptr,   // [1]
    const int*   __restrict__ bidx,     // [B]
    const int*   __restrict__ nnidx,    // [B, NN]
    float* __restrict__ out,            // B*R preds | B vars | R sigma
    float* __restrict__ ws,             // [B, R] sigma partials
    int B)
{
  __shared__ __align__(16) float s_xn[NNB * XNS];   // neighbor features
  __shared__ __align__(16) float s_xb[NF];          // query features
  __shared__ float s_K  [NNB * KS];                 // kernel matrix / Cholesky
  __shared__ float s_rhs[NNB * RS];                 // RHS columns (in-place solve)
  __shared__ float s_sq [NNB];                      // ||xn_n||^2
  __shared__ float s_kc [NNB];                      // Kcross

  const int b   = blockIdx.x;
  const int tid = threadIdx.x;
  const float ls = ls_ptr[0];
  const int* nn = nnidx + b * NNB;

  // ---- phase 1: stage gathers into LDS -------------------------------------
  if (tid < NF) s_xb[tid] = x[(size_t)bidx[b] * NF + tid];
  for (int idx = tid; idx < NNB * (NF / 4); idx += 256) {
    const int row = idx >> 6;        // idx / 64
    const int c4  = idx & 63;
    const float4 v = *(const float4*)(x + (size_t)nn[row] * NF + c4 * 4);
    *(float4*)(&s_xn[row * XNS + c4 * 4]) = v;
  }
  for (int idx = tid; idx < NNB * RR; idx += 256) {
    const int row = idx / RR, r = idx % RR;
    s_rhs[row * RS + r] = tgt[(size_t)b * NNB * RR + idx];
  }
  __syncthreads();

  // ---- phase 2: squared norms + crosswise Matern (float4 LDS reads) --------
  if (tid < NNB) {
    const float4* xr = (const float4*)&s_xn[tid * XNS];
    float a = 0.f;
    for (int k = 0; k < NF / 4; ++k) {
      const float4 v = xr[k];
      a += v.x * v.x + v.y * v.y + v.z * v.z + v.w * v.w;
    }
    s_sq[tid] = a;
  } else if (tid < 2 * NNB) {
    const int n = tid - NNB;
    const float4* xr  = (const float4*)&s_xn[n * XNS];
    const float4* xbv = (const float4*)s_xb;
    float a = 0.f;
    for (int k = 0; k < NF / 4; ++k) {
      const float4 v = xr[k], w = xbv[k];
      const float dx = w.x - v.x, dy = w.y - v.y, dz = w.z - v.z, dw = w.w - v.w;
      a += dx * dx + dy * dy + dz * dz + dw * dw;
    }
    const float d  = sqrtf(fmaxf(a, 0.f));
    const float s  = SQRT3F * d / ls;
    const float kv = (1.f + s) * __expf(-s);
    s_kc[n] = kv;
    s_rhs[n * RS + RR] = kv;           // column 10 = Kcross
  }
  __syncthreads();

  // ---- phase 3: gram via V_WMMA_F32_16X16X4_F32 with A-fragment reuse ------
  // Wave w owns row-block ti = w/2 and column-blocks tj0 = 2*(w%2), tj0+1.
  // A frag loaded once per K-step, feeds two accumulators (2 indep WMMA chains).
  {
    const int wave  = tid >> 5;          // 8 waves
    const int lane  = tid & 31;
    const int lr    = lane & 15;
    const int khalf = (lane >> 4) << 1;  // K offset: 0 (lanes 0-15) / 2 (16-31)
    const float sinv = SQRT3F / ls;

    const int ti  = wave >> 1;           // 0..3
    const int tj0 = (wave & 1) << 1;     // 0 or 2

    const float* arow  = &s_xn[(ti * 16 + lr) * XNS + khalf];
    const float* brow0 = &s_xn[( tj0      * 16 + lr) * XNS + khalf];
    const float* brow1 = &s_xn[((tj0 + 1) * 16 + lr) * XNS + khalf];

    v8f acc0 = {};
    v8f acc1 = {};
    #pragma unroll 8
    for (int kk = 0; kk < NF; kk += 4) {
      const v2f av = *(const v2f*)(arow  + kk);  // A: 16x4 f32 fragment (shared)
      const v2f b0 = *(const v2f*)(brow0 + kk);  // B: 4x16 f32 fragment (tile 0)
      const v2f b1 = *(const v2f*)(brow1 + kk);  // B: 4x16 f32 fragment (tile 1)
      acc0 = __builtin_amdgcn_wmma_f32_16x16x4_f32(
          false, av, false, b0, (short)0, acc0, false, false);
      acc1 = __builtin_amdgcn_wmma_f32_16x16x4_f32(
          false, av, false, b1, (short)0, acc1, false, false);
    }

    // C/D layout: VGPR r -> M = r (lanes 0-15) or r+8 (lanes 16-31), N = lane%16
    const int rbase = ti * 16 + ((lane >> 4) << 3);
    #pragma unroll
    for (int r = 0; r < 8; ++r) {
      const int rowg = rbase + r;
      const float sqr = s_sq[rowg];
      {
        const int colg = tj0 * 16 + lr;
        const float d2 = sqr + s_sq[colg] - 2.f * acc0[r];
        const float d  = sqrtf(fmaxf(d2, 0.f));
        const float s  = sinv * d;
        float kv = (1.f + s) * __expf(-s);
        if (rowg == colg) kv += EPSV;
        s_K[rowg * KS + colg] = kv;
      }
      {
        const int colg = (tj0 + 1) * 16 + lr;
        const float d2 = sqr + s_sq[colg] - 2.f * acc1[r];
        const float d  = sqrtf(fmaxf(d2, 0.f));
        const float s  = sinv * d;
        float kv = (1.f + s) * __expf(-s);
        if (rowg == colg) kv += EPSV;
        s_K[rowg * KS + colg] = kv;
      }
    }
  }
  __syncthreads();

  // ---- phase 4: in-place Cholesky of Kreg (lower triangle) -----------------
  for (int k = 0; k < NNB; ++k) {
    if (tid == 0) s_K[k * KS + k] = sqrtf(s_K[k * KS + k]);
    __syncthreads();
    const float dinv = 1.f / s_K[k * KS + k];
    for (int i = k + 1 + tid; i < NNB; i += 256) s_K[i * KS + k] *= dinv;
    __syncthreads();
    const int rem = NNB - 1 - k;
    for (int idx = tid; idx < rem * rem; idx += 256) {
      const int i = k + 1 + idx / rem;
      const int j = k + 1 + idx % rem;
      if (j <= i) s_K[i * KS + j] -= s_K[i * KS + k] * s_K[j * KS + k];
    }
    __syncthreads();
  }

  // ---- phase 5: triangular solves, 16 lanes per RHS column -----------------
  {
    const int g = tid >> 4;   // column group 0..15 (use 0..10)
    const int l = tid & 15;
    if (g < NCOL) {
      // forward: L y = rhs
      for (int i = 0; i < NNB; ++i) {
        float p = 0.f;
        for (int j = l; j < i; j += 16) p += s_K[i * KS + j] * s_rhs[j * RS + g];
        #pragma unroll
        for (int off = 8; off > 0; off >>= 1) p += __shfl_xor(p, off, 16);
        if (l == 0) s_rhs[i * RS + g] = (s_rhs[i * RS + g] - p) / s_K[i * KS + i];
      }
      // backward: L^T xsol = y
      for (int i = NNB - 1; i >= 0; --i) {
        float p = 0.f;
        for (int j = i + 1 + l; j < NNB; j += 16) p += s_K[j * KS + i] * s_rhs[j * RS + g];
        #pragma unroll
        for (int off = 8; off > 0; off >>= 1) p += __shfl_xor(p, off, 16);
        if (l == 0) s_rhs[i * RS + g] = (s_rhs[i * RS + g] - p) / s_K[i * KS + i];
      }
    }
  }
  __syncthreads();

  // ---- phase 6: outputs ----------------------------------------------------
  if (tid < RR) {
    float p = 0.f, sg = 0.f;
    const float* tb = tgt + (size_t)b * NNB * RR;
    for (int n = 0; n < NNB; ++n) {
      const float ysv = s_rhs[n * RS + tid];
      p  += s_kc[n] * ysv;
      sg += tb[n * RR + tid] * ysv;
    }
    out[(size_t)b * RR + tid] = p;                 // predictions [B,1,R]
    ws [(size_t)b * RR + tid] = sg;                // sigma partial
  } else if (tid == RR) {
    float v = 0.f;
    for (int n = 0; n < NNB; ++n) v += s_kc[n] * s_rhs[n * RS + RR];
    out[(size_t)B * RR + b] = 1.f - v;             // variances [B]
  }
}

// Deterministic fixed-order reduction of sigma partials over batches.
__global__ __launch_bounds__(256) void sigma_reduce_kernel(
    const float* __restrict__ ws, float* __restrict__ out, int B)
{
  __shared__ float red[256];
  const int r = blockIdx.x;           // 0..R-1
  float acc = 0.f;
  for (int b = threadIdx.x; b < B; b += 256) acc += ws[(size_t)b * RR + r];
  red[threadIdx.x] = acc;
  __syncthreads();
  for (int s = 128; s > 0; s >>= 1) {
    if (threadIdx.x < s) red[threadIdx.x] += red[threadIdx.x + s];
    __syncthreads();
  }
  if (threadIdx.x == 0)
    out[(size_t)B * RR + B + r] = red[0] / ((float)NNB * (float)B);
}

extern "C" void kernel_launch(void* const* d_in, const int* in_sizes, int n_in,
                              void* d_out, int out_size, void* d_ws, size_t ws_size,
                              hipStream_t stream) {
  const float* x     = (const float*)d_in[0];
  const float* tgt   = (const float*)d_in[1];
  const float* ls    = (const float*)d_in[2];
  const int*   bidx  = (const int*)  d_in[3];
  const int*   nnidx = (const int*)  d_in[4];
  float* out = (float*)d_out;
  float* ws  = (float*)d_ws;
  const int B = in_sizes[3];

  muygps_batch_kernel<<<B, 256, 0, stream>>>(x, tgt, ls, bidx, nnidx, out, ws, B);
  sigma_reduce_kernel<<<RR, 256, 0, stream>>>(ws, out, B);
}